// EthicalQGANGenerator_33767032881312
// MI455X (gfx1250) — compile-verified
//
#include <hip/hip_runtime.h>

typedef float v2f __attribute__((ext_vector_type(2)));
typedef float v8f __attribute__((ext_vector_type(8)));

#define N_Q 10
#define N_LAYERS 3
#define WAVES_PER_BLOCK 16
#define BLOCK_THREADS 512

// One wave32 per batch element. Amplitude index a[9:0]:
//   a[9:5] = lane id (wires 0..4, since wire w <-> bit 9-w)
//   a[4:0] = register index k (wires 5..9)
__global__ __launch_bounds__(BLOCK_THREADS)
void qgan_circuit_mlp_kernel(const float* __restrict__ z,
                             const float* __restrict__ wts,
                             const float* __restrict__ W1,
                             const float* __restrict__ b1,
                             const float* __restrict__ W2,
                             const float* __restrict__ b2,
                             const float* __restrict__ W3,
                             const float* __restrict__ b3,
                             float* __restrict__ out)
{
    __shared__ float s_feats[16][16];   // per-block 16x(10 pad to 16) feature tile; reused as h2 tile
    __shared__ float s_h1[16][32];      // layer-1 activations staging (D-layout -> A-layout)

    const int tid  = threadIdx.x;
    const int lane = tid & 31;
    const int wv   = tid >> 5;
    const int bidx = blockIdx.x * WAVES_PER_BLOCK + wv;

    // ---------------- RY encoding layer: product state, computed analytically ----------------
    float cw[N_Q], sw[N_Q];
#pragma unroll
    for (int i = 0; i < N_Q; ++i) {
        float zi = z[bidx * N_Q + i];
        zi = fminf(fmaxf(zi, -1.f), 1.f);
        cw[i] = __cosf(0.5f * zi);
        sw[i] = __sinf(0.5f * zi);
    }
    float laneprod = 1.f;
#pragma unroll
    for (int w = 0; w < 5; ++w) {           // wires 0..4 -> lane bit (4-w)
        int lb = 4 - w;
        laneprod *= ((lane >> lb) & 1) ? sw[w] : cw[w];
    }
    float re[32], im[32];
#pragma unroll
    for (int k = 0; k < 32; ++k) {
        float pk = laneprod;
#pragma unroll
        for (int w = 5; w < 10; ++w) {      // wires 5..9 -> reg bit (9-w)
            int bw = 9 - w;
            pk *= ((k >> bw) & 1) ? sw[w] : cw[w];
        }
        re[k] = pk;
        im[k] = 0.f;
    }

    // ---------------- 3 entangling layers: 10 x U3 gates + CNOT chain ----------------
    for (int l = 0; l < N_LAYERS; ++l) {
#pragma unroll
        for (int w = 0; w < N_Q; ++w) {
            const float a  = wts[(l * N_Q + w) * 3 + 0];
            const float bb = wts[(l * N_Q + w) * 3 + 1];
            const float c  = wts[(l * N_Q + w) * 3 + 2];
            const float p  = 0.5f * (a + c);
            const float m  = 0.5f * (a - c);
            const float cb = __cosf(0.5f * bb), sb = __sinf(0.5f * bb);
            const float cp = __cosf(p), sp = __sinf(p);
            const float cm = __cosf(m), sm = __sinf(m);
            // U = [[cb e^{-ip}, -sb e^{+im}], [sb e^{-im}, cb e^{+ip}]]
            const float U00r =  cb * cp, U00i = -cb * sp;
            const float U01r = -sb * cm, U01i = -sb * sm;
            const float U10r =  sb * cm, U10i = -sb * sm;
            const float U11r =  cb * cp, U11i =  cb * sp;

            const int bw = 9 - w;
            if (bw >= 5) {
                // cross-lane wire: lane bit lb
                const int lb  = bw - 5;
                const int bit = (lane >> lb) & 1;
                const float c0r = bit ? U10r : U00r, c0i = bit ? U10i : U00i;
                const float c1r = bit ? U11r : U01r, c1i = bit ? U11i : U01i;
#pragma unroll
                for (int k = 0; k < 32; ++k) {
                    float orr = __shfl_xor(re[k], 1 << lb, 32);
                    float oii = __shfl_xor(im[k], 1 << lb, 32);
                    float a0r = bit ? orr   : re[k], a0i = bit ? oii   : im[k];
                    float a1r = bit ? re[k] : orr,   a1i = bit ? im[k] : oii;
                    re[k] = c0r * a0r - c0i * a0i + c1r * a1r - c1i * a1i;
                    im[k] = c0r * a0i + c0i * a0r + c1r * a1i + c1i * a1r;
                }
            } else {
                // in-register wire: reg bit bw
#pragma unroll
                for (int j = 0; j < 16; ++j) {
                    int lo0 = ((j >> bw) << (bw + 1)) | (j & ((1 << bw) - 1));
                    int lo1 = lo0 | (1 << bw);
                    float x0r = re[lo0], x0i = im[lo0];
                    float x1r = re[lo1], x1i = im[lo1];
                    re[lo0] = U00r * x0r - U00i * x0i + U01r * x1r - U01i * x1i;
                    im[lo0] = U00r * x0i + U00i * x0r + U01r * x1i + U01i * x1r;
                    re[lo1] = U10r * x0r - U10i * x0i + U11r * x1r - U11i * x1i;
                    im[lo1] = U10r * x0i + U10i * x0r + U11r * x1i + U11i * x1r;
                }
            }
        }

        // CNOT chain: ctrl = wire i (bit 9-i), target = wire i+1 (bit 8-i)
#pragma unroll
        for (int i = 0; i < N_Q - 1; ++i) {
            if (i <= 3) {
                // both lane bits: ctrl lane bit (4-i), target lane bit (3-i)
                const int cbit = 4 - i, tbit = 3 - i;
                const int ctrl = (lane >> cbit) & 1;
#pragma unroll
                for (int k = 0; k < 32; ++k) {
                    float tr = __shfl_xor(re[k], 1 << tbit, 32);
                    float ti = __shfl_xor(im[k], 1 << tbit, 32);
                    re[k] = ctrl ? tr : re[k];
                    im[k] = ctrl ? ti : im[k];
                }
            } else if (i == 4) {
                // ctrl = lane bit 0, target = reg bit 4
                const int cond = lane & 1;
#pragma unroll
                for (int lo = 0; lo < 16; ++lo) {
                    float t0r = re[lo], t1r = re[lo | 16];
                    re[lo]      = cond ? t1r : t0r;
                    re[lo | 16] = cond ? t0r : t1r;
                    float t0i = im[lo], t1i = im[lo | 16];
                    im[lo]      = cond ? t1i : t0i;
                    im[lo | 16] = cond ? t0i : t1i;
                }
            } else {
                // both reg bits: ctrl bit (9-i), target bit (8-i) -- pure register swap
                const int cbit = 9 - i, tbit = 8 - i;
#pragma unroll
                for (int j = 0; j < 8; ++j) {
                    int low = j & ((1 << tbit) - 1);
                    int hi  = j >> tbit;
                    int k0  = (hi << (cbit + 1)) | (1 << cbit) | low;
                    int k1  = k0 | (1 << tbit);
                    float t;
                    t = re[k0]; re[k0] = re[k1]; re[k1] = t;
                    t = im[k0]; im[k0] = im[k1]; im[k1] = t;
                }
            }
        }
    }

    // ---------------- Z-expectation features ----------------
    float tot = 0.f;
    float sbits[5] = {0.f, 0.f, 0.f, 0.f, 0.f};
#pragma unroll
    for (int k = 0; k < 32; ++k) {
        float pr = re[k] * re[k] + im[k] * im[k];
        tot += pr;
#pragma unroll
        for (int b = 0; b < 5; ++b)
            if ((k >> b) & 1) sbits[b] += pr;
    }
    float f[N_Q];
#pragma unroll
    for (int w = 0; w < 5; ++w) {           // sign from lane bit (4-w)
        int lb = 4 - w;
        f[w] = ((lane >> lb) & 1) ? -tot : tot;
    }
#pragma unroll
    for (int w = 5; w < 10; ++w)            // sign from reg bit (9-w)
        f[w] = tot - 2.f * sbits[9 - w];

#pragma unroll
    for (int msk = 1; msk < 32; msk <<= 1) {
#pragma unroll
        for (int w = 0; w < N_Q; ++w)
            f[w] += __shfl_xor(f[w], msk, 32);
    }
    if (lane == 0) {
#pragma unroll
        for (int w = 0; w < N_Q; ++w) s_feats[wv][w] = f[w];
#pragma unroll
        for (int w = N_Q; w < 16; ++w) s_feats[wv][w] = 0.f;
    }
    __syncthreads();

    // ---------------- MLP head: wave 0 runs WMMA f32 16x16x4 GEMMs ----------------
    if (wv == 0) {
        const int row  = lane & 15;         // A-frag: lane%16 = M; also B-frag: lane%16 = N
        const int koff = (lane >> 4) << 1;  // lane half selects K pair {0,1} or {2,3}
        const int moff = (lane >> 4) * 8;   // C/D-frag: upper lane half holds M = r+8

        // Layer 1: feats(16x16, K padded) @ W1(10x32) -> two 16x16 N-tiles
        v8f acc0 = {}; v8f acc1 = {};
#pragma unroll
        for (int kk = 0; kk < 4; ++kk) {
            const int k0 = 4 * kk + koff;
            v2f av;  av.x  = s_feats[row][k0];
                     av.y  = s_feats[row][k0 + 1];
            v2f bv0; bv0.x = (k0 < N_Q)     ? W1[k0 * 32 + row]            : 0.f;
                     bv0.y = (k0 + 1 < N_Q) ? W1[(k0 + 1) * 32 + row]      : 0.f;
            v2f bv1; bv1.x = (k0 < N_Q)     ? W1[k0 * 32 + 16 + row]       : 0.f;
                     bv1.y = (k0 + 1 < N_Q) ? W1[(k0 + 1) * 32 + 16 + row] : 0.f;
            acc0 = __builtin_amdgcn_wmma_f32_16x16x4_f32(false, av, false, bv0, (short)0, acc0, false, false);
            acc1 = __builtin_amdgcn_wmma_f32_16x16x4_f32(false, av, false, bv1, (short)0, acc1, false, false);
        }
        // bias + tanh, stage D-layout -> LDS (row-major) for next A-frag
#pragma unroll
        for (int r = 0; r < 8; ++r) {
            const int M = r + moff;
            s_h1[M][row]      = tanhf(acc0[r] + b1[row]);
            s_h1[M][16 + row] = tanhf(acc1[r] + b1[16 + row]);
        }

        // Layer 2: h1(16x32) @ W2(32x16) -> 16x16
        v8f acc2 = {};
#pragma unroll
        for (int kk = 0; kk < 8; ++kk) {
            const int k0 = 4 * kk + koff;
            v2f av; av.x = s_h1[row][k0];
                    av.y = s_h1[row][k0 + 1];
            v2f bv; bv.x = W2[k0 * 16 + row];
                    bv.y = W2[(k0 + 1) * 16 + row];
            acc2 = __builtin_amdgcn_wmma_f32_16x16x4_f32(false, av, false, bv, (short)0, acc2, false, false);
        }
#pragma unroll
        for (int r = 0; r < 8; ++r) {
            const int M = r + moff;
            s_feats[M][row] = tanhf(acc2[r] + b2[row]);   // reuse s_feats as h2 tile
        }

        // Layer 3: h2(16x16) @ W3(16x1) + b3 -> sigmoid * 0.2 - 0.1
        if (lane < 16) {
            float raw = b3[0];
#pragma unroll
            for (int n = 0; n < 16; ++n)
                raw += s_feats[lane][n] * W3[n];
            float sg = 1.f / (1.f + __expf(-raw));
            out[blockIdx.x * WAVES_PER_BLOCK + lane] = sg * 0.2f - 0.1f;
        }
    }
}

extern "C" void kernel_launch(void* const* d_in, const int* in_sizes, int n_in,
                              void* d_out, int out_size, void* d_ws, size_t ws_size,
                              hipStream_t stream) {
    (void)in_sizes; (void)n_in; (void)d_ws; (void)ws_size;
    const float* z   = (const float*)d_in[0];
    const float* wts = (const float*)d_in[1];
    const float* W1  = (const float*)d_in[2];
    const float* b1  = (const float*)d_in[3];
    const float* W2  = (const float*)d_in[4];
    const float* b2  = (const float*)d_in[5];
    const float* W3  = (const float*)d_in[6];
    const float* b3  = (const float*)d_in[7];
    float* out = (float*)d_out;

    const int nblocks = out_size / WAVES_PER_BLOCK;   // 4096 / 16 = 256
    qgan_circuit_mlp_kernel<<<nblocks, BLOCK_THREADS, 0, stream>>>(
        z, wts, W1, b1, W2, b2, W3, b3, out);
}